// GCN_48206712930318
// MI455X (gfx1250) — compile-verified
//
#include <hip/hip_runtime.h>
#include <hip/hip_bf16.h>
#include <math.h>

// ---- problem constants (match reference) ----
#define NNODES   100000
#define NEDGES   3200000
#define F_IN     512
#define HID      256
#define NCLS     40

typedef float v2f __attribute__((ext_vector_type(2)));
typedef float v8f __attribute__((ext_vector_type(8)));

// ---------------------------------------------------------------------------
// Dense GEMM: C[M,N] = A[M,K] * B[K,N], f32 via V_WMMA_F32_16X16X4_F32.
// Block = 256 threads (8 wave32s). Block tile 128(M) x 32(N).
// Wave (wm in 0..3, wn in 0..1) owns a 32x16 region = 2 accumulators.
// K stepped by 16 (4 wmma k-steps of 4) through LDS.
// Requires K % 16 == 0 and lda % 4 == 0 (true: K=512/256, lda=512/256).
// ---------------------------------------------------------------------------
#define AS_STRIDE 20   // 16 cols padded to 20 floats: 16B aligned, bank-conflict-free

__global__ __launch_bounds__(256)
void gemm_f32_wmma(const float* __restrict__ A, const float* __restrict__ B,
                   float* __restrict__ C, int M, int N, int K,
                   int lda, int ldb, int ldc) {
  __shared__ float As[128 * AS_STRIDE];
  __shared__ float Bs[16 * 32];

  const int tid  = threadIdx.x;
  const int lane = tid & 31;
  const int wave = tid >> 5;
  const int wm   = wave & 3;   // M sub-tile (32 rows each)
  const int wn   = wave >> 2;  // N sub-tile (16 cols each)
  const int mn   = lane & 15;  // row (A) / col (B,C) within 16
  const int half = lane >> 4;  // half-wave selects K pair
  const int koff = half * 2;

  const int rowBase = blockIdx.x * 128;
  const int nBase   = blockIdx.y * 32;

  v8f acc0 = {0.f, 0.f, 0.f, 0.f, 0.f, 0.f, 0.f, 0.f};
  v8f acc1 = acc0;

  for (int kt = 0; kt < K; kt += 16) {
    // --- cooperative load A tile: 128x16 floats, float4 per thread x2 ---
#pragma unroll
    for (int i = 0; i < 2; ++i) {
      int idx4 = tid + i * 256;         // 0..511 float4 slots
      int r    = idx4 >> 2;             // 0..127
      int c4   = (idx4 & 3) << 2;       // 0,4,8,12
      int gr   = rowBase + r;
      float4 v = make_float4(0.f, 0.f, 0.f, 0.f);
      if (gr < M) v = *(const float4*)(A + (size_t)gr * lda + kt + c4);
      As[r * AS_STRIDE + c4 + 0] = v.x;
      As[r * AS_STRIDE + c4 + 1] = v.y;
      As[r * AS_STRIDE + c4 + 2] = v.z;
      As[r * AS_STRIDE + c4 + 3] = v.w;
    }
    // --- cooperative load B tile: 16x32 floats, guarded for N edge ---
#pragma unroll
    for (int i = 0; i < 2; ++i) {
      int idx = tid + i * 256;          // 0..511
      int r   = idx >> 5;               // 0..15
      int c   = idx & 31;
      int gc  = nBase + c;
      Bs[r * 32 + c] = (gc < N) ? B[(size_t)(kt + r) * ldb + gc] : 0.f;
    }
    __syncthreads();

    const int ar0 = (wm * 32 + mn) * AS_STRIDE;
    const int ar1 = ar0 + 16 * AS_STRIDE;
    const int bc  = wn * 16 + mn;
#pragma unroll
    for (int k4 = 0; k4 < 16; k4 += 4) {
      const int kk = k4 + koff;
      v2f a0 = {As[ar0 + kk], As[ar0 + kk + 1]};
      v2f a1 = {As[ar1 + kk], As[ar1 + kk + 1]};
      v2f bb = {Bs[kk * 32 + bc], Bs[(kk + 1) * 32 + bc]};
      acc0 = __builtin_amdgcn_wmma_f32_16x16x4_f32(
          false, a0, false, bb, (short)0, acc0, false, false);
      acc1 = __builtin_amdgcn_wmma_f32_16x16x4_f32(
          false, a1, false, bb, (short)0, acc1, false, false);
    }
    __syncthreads();
  }

  // C/D layout: VGPR r, lanes 0-15 -> M=r, lanes 16-31 -> M=r+8; N = lane&15
  const int gc = nBase + wn * 16 + mn;
#pragma unroll
  for (int r = 0; r < 8; ++r) {
    int gr0 = rowBase + wm * 32 + r + 8 * half;
    if (gr0 < M && gc < N) C[(size_t)gr0 * ldc + gc] = acc0[r];
    int gr1 = gr0 + 16;
    if (gr1 < M && gc < N) C[(size_t)gr1 * ldc + gc] = acc1[r];
  }
}

// ---------------------------------------------------------------------------
// Edge-parallel SpMM scatter-add: dst[row[e]] += val[e] * src[col[e]]
// One thread per (edge, feature). Gather rows + accumulator rows live in L2
// (support matrix is 102.4MB < 192MB L2), atomics resolve at L2.
// ---------------------------------------------------------------------------
template <int F>
__global__ __launch_bounds__(256)
void spmm_atomic(const int* __restrict__ erow, const int* __restrict__ ecol,
                 const float* __restrict__ eval, const float* __restrict__ src,
                 float* __restrict__ dst, long long total) {
  long long idx = (long long)blockIdx.x * 256 + threadIdx.x;
  if (idx >= total) return;
  int e = (int)(idx / F);
  int f = (int)(idx % F);
  int r = erow[e];
  int c = ecol[e];
  float v = eval[e] * src[(size_t)c * F + f];
  atomicAdd(dst + (size_t)r * F + f, v);
}

__global__ __launch_bounds__(256)
void zero_f32(float* __restrict__ p, long long n) {
  long long i = (long long)blockIdx.x * 256 + threadIdx.x;
  if (i < n) p[i] = 0.f;
}

__global__ __launch_bounds__(256)
void bias_relu(float* __restrict__ h, const float* __restrict__ b,
               long long n, int F) {
  long long i = (long long)blockIdx.x * 256 + threadIdx.x;
  if (i >= n) return;
  float v = h[i] + b[(int)(i % F)];
  h[i] = v > 0.f ? v : 0.f;
}

// ---------------------------------------------------------------------------
// Fused bias + log_softmax + softmax epilogue. One thread per row (C=40).
// Output: [0, 4M) = log_softmax, [4M, 8M) = softmax.
// ---------------------------------------------------------------------------
__global__ __launch_bounds__(256)
void softmax_out(const float* __restrict__ logits, const float* __restrict__ b,
                 float* __restrict__ out, int n) {
  int i = blockIdx.x * 256 + threadIdx.x;
  if (i >= n) return;
  float l[NCLS];
  float m = -INFINITY;
#pragma unroll
  for (int j = 0; j < NCLS; ++j) {
    l[j] = logits[(size_t)i * NCLS + j] + b[j];
    m = fmaxf(m, l[j]);
  }
  float s = 0.f;
#pragma unroll
  for (int j = 0; j < NCLS; ++j) s += expf(l[j] - m);
  float logs = logf(s);
  float inv  = 1.f / s;
  const size_t off = (size_t)NNODES * NCLS;
#pragma unroll
  for (int j = 0; j < NCLS; ++j) {
    float t = l[j] - m;
    out[(size_t)i * NCLS + j]       = t - logs;        // log_softmax
    out[off + (size_t)i * NCLS + j] = expf(t) * inv;   // softmax
  }
}

// ---------------------------------------------------------------------------
extern "C" void kernel_launch(void* const* d_in, const int* in_sizes, int n_in,
                              void* d_out, int out_size, void* d_ws, size_t ws_size,
                              hipStream_t stream) {
  const float* x    = (const float*)d_in[0];
  const float* W1   = (const float*)d_in[1];
  const float* b1   = (const float*)d_in[2];
  const float* W2   = (const float*)d_in[3];
  const float* b2   = (const float*)d_in[4];
  const int*   erow = (const int*)d_in[5];
  const int*   ecol = (const int*)d_in[6];
  const float* evalp= (const float*)d_in[7];
  float* out = (float*)d_out;

  // workspace layout (floats):
  //   bufA [25.6M]  : support1 (100k x 256); later reused:
  //                   [0,4M)  -> support2 (100k x 40)
  //                   [4M,8M) -> logits accumulator (100k x 40)
  //   bufH [25.6M]  : layer-1 spmm accumulator -> h (in place)
  float* bufA     = (float*)d_ws;
  float* bufH     = bufA + (size_t)NNODES * HID;       // +25.6M floats
  float* support2 = bufA;
  float* logits   = bufA + (size_t)NNODES * NCLS;      // +4M floats

  const long long nH = (long long)NNODES * HID;        // 25.6M
  const long long nL = (long long)NNODES * NCLS;       // 4M

  // 1) zero layer-1 accumulator
  zero_f32<<<(unsigned)((nH + 255) / 256), 256, 0, stream>>>(bufH, nH);

  // 2) support1 = x @ W1   [100000 x 256]
  gemm_f32_wmma<<<dim3((NNODES + 127) / 128, (HID + 31) / 32), 256, 0, stream>>>(
      x, W1, bufA, NNODES, HID, F_IN, F_IN, HID, HID);

  // 3) spmm layer 1: bufH[row] += val * support1[col]
  {
    long long total = (long long)NEDGES * HID;         // 819.2M
    spmm_atomic<HID><<<(unsigned)((total + 255) / 256), 256, 0, stream>>>(
        erow, ecol, evalp, bufA, bufH, total);
  }

  // 4) h = relu(acc + b1), in place
  bias_relu<<<(unsigned)((nH + 255) / 256), 256, 0, stream>>>(bufH, b1, nH, HID);

  // 5) support2 = h @ W2   [100000 x 40]
  gemm_f32_wmma<<<dim3((NNODES + 127) / 128, (NCLS + 31) / 32), 256, 0, stream>>>(
      bufH, W2, support2, NNODES, NCLS, HID, HID, NCLS, NCLS);

  // 6) zero logits accumulator
  zero_f32<<<(unsigned)((nL + 255) / 256), 256, 0, stream>>>(logits, nL);

  // 7) spmm layer 2
  {
    long long total = (long long)NEDGES * NCLS;        // 128M
    spmm_atomic<NCLS><<<(unsigned)((total + 255) / 256), 256, 0, stream>>>(
        erow, ecol, evalp, support2, logits, total);
  }

  // 8) bias + log_softmax + softmax -> d_out
  softmax_out<<<(NNODES + 255) / 256, 256, 0, stream>>>(logits, b2, out, NNODES);
}